// BatchPooling_1821066134188
// MI455X (gfx1250) — compile-verified
//
#include <hip/hip_runtime.h>
#include <math.h>
#include <stdint.h>

// Segment-max pooling: out[b][d] = max over rows r with batch[r]==b of x[r][d].
// N=1M rows, D=128, B=2048 sorted segments. Pure HBM-streaming workload:
// 512 MB read @ 23.3 TB/s => ~22us floor. Strategy: async global->LDS b128
// copies (ASYNCcnt-pipelined, non-temporal) + v_max reduction.

#define WG        256
#define TILE_ROWS 8      // rows per tile (one row per wave32)
#define NBUF      4      // LDS double^2 buffering
#define PD        3      // async issue depth (NBUF-1)
#define DFIX      128    // feature dim this kernel is specialized for

__device__ __forceinline__ int lower_bound_i32(const int* __restrict__ a,
                                               int n, int key) {
  int lo = 0, hi = n;
  while (lo < hi) {
    int mid = (lo + hi) >> 1;
    if (a[mid] < key) lo = mid + 1; else hi = mid;
  }
  return lo;
}

__global__ __launch_bounds__(WG) void seg_max_d128(
    const float* __restrict__ x, const int* __restrict__ batch,
    float* __restrict__ out, int N) {
  __shared__ float4 lds[NBUF * WG];   // 16 KB: NBUF tiles of 8 rows x 128 f32

  const int b    = blockIdx.x;
  const int tid  = (int)threadIdx.x;
  const int c4   = tid & 31;          // float4 column index (lane within wave)
  const int rofs = tid >> 5;          // row-slice within tile (wave id, 0..7)

  // Segment boundaries via binary search on sorted batch ids (uniform per block).
  const int r0   = lower_bound_i32(batch, N, b);
  const int r1   = lower_bound_i32(batch, N, b + 1);
  const int rows = r1 - r0;

  float4 acc;
  acc.x = acc.y = acc.z = acc.w = -INFINITY;

  if (rows > 0) {
    const int T    = (rows + TILE_ROWS - 1) / TILE_ROWS;
    const int rmax = r1 - 1;
    // Wave-relative LDS byte address of this thread's 16B slot in buffer 0.
    const unsigned lds_slot =
        (unsigned)(uintptr_t)(&lds[0]) + (unsigned)tid * 16u;

    // ---- prologue: prime the async pipeline (PD tiles in flight) ----
#pragma unroll
    for (int t = 0; t < PD; ++t) {
      int row = r0 + t * TILE_ROWS + rofs;
      if (row > rmax) row = rmax;                       // clamp: dup rows ok for max
      unsigned goff  = (unsigned)row * (DFIX * 4u) + (unsigned)c4 * 16u;
      unsigned laddr = lds_slot + (unsigned)(t & (NBUF - 1)) * (WG * 16u);
      asm volatile(
          "global_load_async_to_lds_b128 %0, %1, %2 th:TH_LOAD_NT"
          :: "v"(laddr), "v"(goff), "s"(x) : "memory");
    }

    // ---- steady state: always keep exactly PD newer async ops in flight ----
    for (int t = 0; t < T; ++t) {
      {
        const int tn = t + PD;                           // may exceed T: clamped dup
        int row = r0 + tn * TILE_ROWS + rofs;
        if (row > rmax) row = rmax;
        unsigned goff  = (unsigned)row * (DFIX * 4u) + (unsigned)c4 * 16u;
        unsigned laddr = lds_slot + (unsigned)(tn & (NBUF - 1)) * (WG * 16u);
        asm volatile(
            "global_load_async_to_lds_b128 %0, %1, %2 th:TH_LOAD_NT"
            :: "v"(laddr), "v"(goff), "s"(x) : "memory");
      }
      // 3 newer async ops were issued after tile t; in-order completion =>
      // ASYNCcnt<=3 guarantees tile t's data landed in LDS.
      asm volatile("s_wait_asynccnt 3" ::: "memory");

      float4 v = lds[(t & (NBUF - 1)) * WG + tid];       // ds_load_b128, own slot
      acc.x = fmaxf(acc.x, v.x);
      acc.y = fmaxf(acc.y, v.y);
      acc.z = fmaxf(acc.z, v.z);
      acc.w = fmaxf(acc.w, v.w);
    }
  }

  // ---- cross-wave reduction over the 8 row slices ----
  asm volatile("s_wait_asynccnt 0" ::: "memory");  // drain before reusing LDS
  __syncthreads();
  lds[tid] = acc;                                  // [rofs*32 + c4]
  __syncthreads();
  if (tid < 32) {
    float4 m = lds[tid];
#pragma unroll
    for (int w = 1; w < TILE_ROWS; ++w) {
      float4 v = lds[w * 32 + tid];
      m.x = fmaxf(m.x, v.x);
      m.y = fmaxf(m.y, v.y);
      m.z = fmaxf(m.z, v.z);
      m.w = fmaxf(m.w, v.w);
    }
    ((float4*)out)[(size_t)b * (DFIX / 4) + tid] = m;  // global_store_b128
  }
}

// Fallback for D != 128: one thread per output element, sequential row scan.
__global__ void seg_max_generic(const float* __restrict__ x,
                                const int* __restrict__ batch,
                                float* __restrict__ out,
                                int N, int D, int B) {
  int i = (int)(blockIdx.x * blockDim.x + threadIdx.x);
  if (i >= B * D) return;
  int b = i / D, d = i - b * D;
  int r0 = lower_bound_i32(batch, N, b);
  int r1 = lower_bound_i32(batch, N, b + 1);
  float m = -INFINITY;
  for (int r = r0; r < r1; ++r) m = fmaxf(m, x[(size_t)r * D + d]);
  out[i] = m;
}

extern "C" void kernel_launch(void* const* d_in, const int* in_sizes, int n_in,
                              void* d_out, int out_size, void* d_ws, size_t ws_size,
                              hipStream_t stream) {
  (void)n_in; (void)d_ws; (void)ws_size;
  const float* x     = (const float*)d_in[0];
  const int*   batch = (const int*)d_in[1];
  float*       out   = (float*)d_out;

  const int N = in_sizes[1];
  const int D = in_sizes[0] / in_sizes[1];
  const int B = out_size / D;

  if (D == DFIX) {
    seg_max_d128<<<B, WG, 0, stream>>>(x, batch, out, N);
  } else {
    int tot = B * D;
    seg_max_generic<<<(tot + 255) / 256, 256, 0, stream>>>(x, batch, out, N, D, B);
  }
}